// StablePolicy3phase_79963701117394
// MI455X (gfx1250) — compile-verified
//
#include <hip/hip_runtime.h>

typedef __attribute__((ext_vector_type(2))) float v2f;
typedef __attribute__((ext_vector_type(8))) float v8f;

#define VMAXC 1.05f
#define VMINC 0.95f
#define SCALEC 0.15f

__device__ __forceinline__ float deadband(float x) {
    // s = relu(x - (VMAX-0.02)) - relu((VMIN+0.02) - x)
    return fmaxf(x - (VMAXC - 0.02f), 0.0f) - fmaxf((VMINC + 0.02f) - x, 0.0f);
}

// ---------------------------------------------------------------------------
// Prep kernel: one block of 256 threads (H = 256 heads).
//  - sumabs = sum(relu(b)) over all 768 entries (LDS tree reduction)
//  - bn'[e*256+h] = SCALE * relu(b[h*3+e]) / sumabs   (e-major permutation)
//  - W[h] = sum_k lambda[h,k]^2 * E[k]  written directly as WMMA B-operand
//    fragments: column c = e*256+h -> group g = c/16, N = c%16.
//    frag[g*64 + lane*2 + v]: lane<16 holds {K0,K1}, lane>=16 holds {K2, 0}.
// ---------------------------------------------------------------------------
__global__ __launch_bounds__(256) void prep_kernel(
    const float* __restrict__ b,          // [256*3]
    const float* __restrict__ lambda_pst, // [256*6]
    float* __restrict__ frag,             // [48*64] = 3072
    float* __restrict__ bnp)              // [768]
{
    __shared__ float red[256];
    const int t = threadIdx.x; // == head h
    float partial = 0.0f;
    for (int i = t; i < 768; i += 256) partial += fmaxf(b[i], 0.0f);
    red[t] = partial;
    __syncthreads();
    for (int s = 128; s > 0; s >>= 1) {
        if (t < s) red[t] += red[t + s];
        __syncthreads();
    }
    const float inv = SCALEC / red[0];

    float l[6];
#pragma unroll
    for (int k = 0; k < 6; ++k) { float v = lambda_pst[t * 6 + k]; l[k] = v * v; }
    // basis order: (0,0),(1,0),(1,1),(2,0),(2,1),(2,2)
    float W[3][3];
    W[0][0] = l[0] + l[1] + l[3]; W[0][1] = -l[1];               W[0][2] = -l[3];
    W[1][0] = -l[1];              W[1][1] = l[1] + l[2] + l[4];  W[1][2] = -l[4];
    W[2][0] = -l[3];              W[2][1] = -l[4];               W[2][2] = l[3] + l[4] + l[5];

    const int hdiv = t >> 4, N = t & 15;
#pragma unroll
    for (int e = 0; e < 3; ++e) {
        const int g = e * 16 + hdiv;
        frag[g * 64 + N * 2 + 0]        = W[0][e]; // K=0
        frag[g * 64 + N * 2 + 1]        = W[1][e]; // K=1
        frag[g * 64 + (N + 16) * 2 + 0] = W[2][e]; // K=2
        frag[g * 64 + (N + 16) * 2 + 1] = 0.0f;    // K=3 pad
        bnp[e * 256 + t] = fmaxf(b[t * 3 + e], 0.0f) * inv;
    }
}

// ---------------------------------------------------------------------------
// Main kernel: 256 threads = 8 waves; each wave owns a 16-row batch tile.
// ---------------------------------------------------------------------------
__global__ __launch_bounds__(256) void main_kernel(
    const float* __restrict__ state,  // [Bn*3]
    const float* __restrict__ frag_g, // [3072]
    const float* __restrict__ bnp_g,  // [768]
    float* __restrict__ out,          // [Bn*3]
    int Bn)
{
    __shared__ float sfrag[3072];
    __shared__ float sbn[768];
    __shared__ float sacc[8 * 32 * 24]; // per-(wave,lane) partials: [e][v]

    const int t = threadIdx.x;
    for (int i = t; i < 3072; i += 256) sfrag[i] = frag_g[i];
    for (int i = t; i < 768;  i += 256) sbn[i]  = bnp_g[i];
    __syncthreads();

    const int wave  = t >> 5;
    const int lane  = t & 31;
    const int laneN = lane & 15;
    const int half  = lane >> 4;

    const long tilebase = (long)blockIdx.x * 128 + wave * 16;

    // A fragment (f32 16x4): lane<16 -> M=lane, {K0,K1}; lane>=16 -> M=lane-16, {K2, 0}
    long arow = tilebase + laneN;
    if (arow >= Bn) arow = Bn - 1;           // clamp loads, keep EXEC all-1s
    const float* srow = state + arow * 3;
    v2f a;
    a.x = deadband(srow[half * 2]);          // K0 or K2
    a.y = (half == 0) ? deadband(srow[1]) : 0.0f;

    float acc[3][8];
#pragma unroll
    for (int e = 0; e < 3; ++e)
#pragma unroll
        for (int v = 0; v < 8; ++v) acc[e][v] = 0.0f;

#pragma unroll
    for (int e = 0; e < 3; ++e) {
#pragma unroll 4
        for (int gg = 0; gg < 16; ++gg) {
            const int g = e * 16 + gg;
            v2f bf;
            bf.x = sfrag[g * 64 + lane * 2 + 0];
            bf.y = sfrag[g * 64 + lane * 2 + 1];
            const float bias = sbn[g * 16 + laneN];
            v8f c = {};
            c = __builtin_amdgcn_wmma_f32_16x16x4_f32(
                    /*neg_a=*/false, a, /*neg_b=*/false, bf,
                    /*c_mod=*/(short)0, c, /*reuse_a=*/false, /*reuse_b=*/false);
#pragma unroll
            for (int v = 0; v < 8; ++v) {
                const float x = c[v];
                const float shr = fmaxf(fabsf(x) - bias, 0.0f);
                acc[e][v] += copysignf(shr, x); // sign(x)*max(|x|-c,0)
            }
        }
    }

    // spill per-lane partials, reduce across the 16 lanes that share a row-half
#pragma unroll
    for (int e = 0; e < 3; ++e)
#pragma unroll
        for (int v = 0; v < 8; ++v)
            sacc[(wave * 32 + lane) * 24 + e * 8 + v] = acc[e][v];
    __syncthreads();

    // 384 outputs per block: e-major tasks
    for (int task = t; task < 384; task += 256) {
        const int e  = task >> 7;       // 0..2
        const int r  = task & 127;      // row within block
        const int w  = r >> 4;
        const int mm = r & 15;
        const int hh = mm >> 3;         // C layout: M<8 -> lanes 0-15, else 16-31
        const int v  = mm & 7;
        const float* base = &sacc[(w * 32 + hh * 16) * 24 + e * 8 + v];
        float s = 0.0f;
#pragma unroll
        for (int ln = 0; ln < 16; ++ln) s += base[ln * 24];
        const long row = (long)blockIdx.x * 128 + r;
        if (row < Bn) {
            const float sv = deadband(state[row * 3 + e]);
            out[row * 3 + e] = s + 0.001f * sv;
        }
    }
}

extern "C" void kernel_launch(void* const* d_in, const int* in_sizes, int n_in,
                              void* d_out, int out_size, void* d_ws, size_t ws_size,
                              hipStream_t stream) {
    (void)n_in; (void)out_size; (void)ws_size;
    const float* state = (const float*)d_in[0]; // [B,3]
    const float* b     = (const float*)d_in[1]; // [256,3]
    const float* lam   = (const float*)d_in[2]; // [256,6]
    float* out = (float*)d_out;

    float* frag = (float*)d_ws;   // 3072 floats
    float* bnp  = frag + 3072;    // 768 floats

    const int Bn = in_sizes[0] / 3;

    prep_kernel<<<1, 256, 0, stream>>>(b, lam, frag, bnp);

    const int blocks = (Bn + 127) / 128;
    main_kernel<<<blocks, 256, 0, stream>>>(state, frag, bnp, out, Bn);
}